// RGCN_32598801776724
// MI455X (gfx1250) — compile-verified
//
#include <hip/hip_runtime.h>

typedef __attribute__((ext_vector_type(2))) float v2f;
typedef __attribute__((ext_vector_type(4))) float v4f;
typedef __attribute__((ext_vector_type(8))) float v8f;

#define N_NODES   100000
#define N_EDGES   1600000
#define NUM_RELS  8
#define NUM_BASES 8
#define E_DIM     32
#define MAX_LEN   8
#define H_DIM     64
#define OUT_DIM   64
#define K_IN      (E_DIM * MAX_LEN)        // 256
#define PERM_CAP  (N_EDGES + 128)          // room for per-relation pad-to-16
#define NTILES    (PERM_CAP / 16)          // 100008

// ---------------------------------------------------------------------------
// Wt[r][o][i] = sum_b comp[r,b] * V[b][i][o]   (stored TRANSPOSED: [8][64][64])
// Transposed layout makes the WMMA B-operand fetch a contiguous b64 load:
// B[k][n] lives at Wt[r][n*64 + k], so (k, k+1) are adjacent in memory.
// ---------------------------------------------------------------------------
__global__ void compute_w_kernel(const float* __restrict__ V,
                                 const float* __restrict__ comp,
                                 float* __restrict__ Wt) {
    const int r = blockIdx.x;
    float c[NUM_BASES];
#pragma unroll
    for (int b = 0; b < NUM_BASES; ++b) c[b] = comp[r * NUM_BASES + b];
    for (int idx = threadIdx.x; idx < H_DIM * H_DIM; idx += blockDim.x) {
        const int i = idx >> 6;            // input dim
        const int o = idx & 63;            // output dim
        float acc = 0.f;
#pragma unroll
        for (int b = 0; b < NUM_BASES; ++b)
            acc += c[b] * V[b * H_DIM * H_DIM + idx];
        Wt[r * H_DIM * H_DIM + o * H_DIM + i] = acc;
    }
}

// ---------------------------------------------------------------------------
// x[n, :] = (emb_w[feat[n]].flatten() @ sm_w.T) + sm_b      via WMMA f32
// One block = 16 nodes; 4 waves each own one 16-column tile; K = 256.
// ---------------------------------------------------------------------------
__global__ __launch_bounds__(128)
void size_matcher_kernel(const int* __restrict__ feat,
                         const float* __restrict__ emb_w,
                         const float* __restrict__ sm_w,
                         const float* __restrict__ sm_b,
                         float* __restrict__ x) {
    __shared__ float A[16 * 264];          // 16 rows, K=256, pad stride 264
    const int node0 = blockIdx.x * 16;

    // Cooperative gather: 128 threads = 16 nodes x 8 slots, 32 floats each.
    {
        const int t    = threadIdx.x;
        const int row  = t >> 3;
        const int slot = t & 7;
        const int idx  = feat[(node0 + row) * MAX_LEN + slot];
        const float* srcp = emb_w + (long)idx * E_DIM;
        float*       dstp = A + row * 264 + slot * E_DIM;
#pragma unroll
        for (int j = 0; j < E_DIM; j += 4)
            *(v4f*)(dstp + j) = *(const v4f*)(srcp + j);
    }
    __syncthreads();

    const int wave = threadIdx.x >> 5;     // column tile 0..3
    const int lane = threadIdx.x & 31;
    const int m    = lane & 15;            // A row / B-&-D column index
    const int kHi  = lane >> 4;            // lanes 16..31 carry K+2,K+3
    const int col  = wave * 16 + m;

    const float* brow = sm_w + (long)col * K_IN;   // B[k][col] = sm_w[col*256+k]
    v8f acc = {};
#pragma unroll 8
    for (int k0 = 0; k0 < K_IN; k0 += 4) {
        const int k = k0 + 2 * kHi;
        const v2f a = *(const v2f*)(A + m * 264 + k);
        const v2f b = *(const v2f*)(brow + k);
        acc = __builtin_amdgcn_wmma_f32_16x16x4_f32(false, a, false, b,
                                                    (short)0, acc, false, false);
    }
    const float bias = sm_b[col];
#pragma unroll
    for (int v = 0; v < 8; ++v) {
        const int row = v + 8 * kHi;       // D layout: VGPR v -> M = v + 8*(lane>=16)
        x[(long)(node0 + row) * H_DIM + col] = acc[v] + bias;
    }
}

// ---------------------------------------------------------------------------
// Relation bucketing: histogram -> padded prefix -> scatter
// ---------------------------------------------------------------------------
__global__ void init_sort_kernel(int* __restrict__ perm,
                                 int* __restrict__ counts, int cap) {
    const int i = blockIdx.x * blockDim.x + threadIdx.x;
    if (i < cap) perm[i] = -1;
    if (i < NUM_RELS) counts[i] = 0;
}

__global__ void hist_kernel(const int* __restrict__ etype,
                            int* __restrict__ counts) {
    const int e = blockIdx.x * blockDim.x + threadIdx.x;
    if (e < N_EDGES) atomicAdd(&counts[etype[e]], 1);
}

__global__ void prefix_kernel(const int* __restrict__ counts,
                              int* __restrict__ cursor) {
    if (threadIdx.x == 0 && blockIdx.x == 0) {
        int run = 0;
        for (int r = 0; r < NUM_RELS; ++r) {
            cursor[r] = run;
            run += (counts[r] + 15) & ~15;   // pad each segment to tile size
        }
    }
}

__global__ void scatter_kernel(const int* __restrict__ etype,
                               int* __restrict__ cursor,
                               int* __restrict__ perm) {
    const int e = blockIdx.x * blockDim.x + threadIdx.x;
    if (e < N_EDGES) {
        const int pos = atomicAdd(&cursor[etype[e]], 1);
        perm[pos] = e;
    }
}

__global__ void init_bias_kernel(float* __restrict__ dst,
                                 const float* __restrict__ bias) {
    const int i = blockIdx.x * blockDim.x + threadIdx.x;   // < N_NODES*64
    dst[i] = bias[i & (H_DIM - 1)];
}

// ---------------------------------------------------------------------------
// One RGCN layer, edge-side: each wave owns a 16-edge single-relation tile.
//   msg = (relu?(x[src]) * norm) @ W[r]   -> atomic scatter into agg[dst]
// Wt is the transposed relation weight: B[k][n] = Wt[r][n*64 + k].
// ---------------------------------------------------------------------------
template <bool RELU_IN>
__global__ __launch_bounds__(128)
void rgcn_layer_kernel(const float* __restrict__ xin,
                       const float* __restrict__ Wt,    // [8][64][64] transposed
                       const int*   __restrict__ perm,
                       const int*   __restrict__ src,
                       const int*   __restrict__ dst,
                       const int*   __restrict__ etype,
                       const float* __restrict__ norm,
                       float*       __restrict__ agg,
                       int ntiles) {
    const int wave = threadIdx.x >> 5;
    const int lane = threadIdx.x & 31;
    const int m    = lane & 15;
    const int kHi  = lane >> 4;
    const int tile = blockIdx.x * 4 + wave;
    if (tile >= ntiles) return;                 // wave-uniform
    const int tbase = tile * 16;
    const int e0 = perm[tbase];
    if (e0 < 0) return;                         // fully padded tile, wave-uniform

    const int r = etype[e0];                    // single relation per tile
    const float* Wr = Wt + (long)r * H_DIM * H_DIM;

    // This lane's A half-row (row m, K-half kHi), with norm (and ReLU) folded in.
    const int   em  = perm[tbase + m];
    const int   sn  = (em >= 0) ? src[em]  : 0;
    const float nrm = (em >= 0) ? norm[em] : 0.f;   // zero row for pad slots
    const float* xrow = xin + (long)sn * H_DIM;
    v2f areg[16];
#pragma unroll
    for (int kk = 0; kk < 16; ++kk) {
        const int k = kk * 4 + 2 * kHi;
        v2f a = *(const v2f*)(xrow + k);
        if (RELU_IN) { a.x = fmaxf(a.x, 0.f); a.y = fmaxf(a.y, 0.f); }
        areg[kk].x = a.x * nrm;
        areg[kk].y = a.y * nrm;
    }

    // Destination rows this lane commits (D rows v + 8*kHi).
    int drow[8];
#pragma unroll
    for (int v = 0; v < 8; ++v) {
        const int er = perm[tbase + v + 8 * kHi];
        drow[v] = (er >= 0) ? dst[er] : -1;
    }

#pragma unroll
    for (int c = 0; c < 4; ++c) {
        v8f acc = {};
        const float* bcol = Wr + (long)(c * 16 + m) * H_DIM;  // contiguous in k
#pragma unroll
        for (int kk = 0; kk < 16; ++kk) {
            const int k = kk * 4 + 2 * kHi;
            const v2f b = *(const v2f*)(bcol + k);            // single b64 load
            acc = __builtin_amdgcn_wmma_f32_16x16x4_f32(false, areg[kk], false, b,
                                                        (short)0, acc, false, false);
        }
#pragma unroll
        for (int v = 0; v < 8; ++v) {
            if (drow[v] >= 0)
                atomicAdd(&agg[(long)drow[v] * H_DIM + c * 16 + m], acc[v]);
        }
    }
}

// ---------------------------------------------------------------------------
extern "C" void kernel_launch(void* const* d_in, const int* in_sizes, int n_in,
                              void* d_out, int out_size, void* d_ws, size_t ws_size,
                              hipStream_t stream) {
    const int*   feat  = (const int*)  d_in[0];
    const int*   src   = (const int*)  d_in[1];
    const int*   dst   = (const int*)  d_in[2];
    const int*   etype = (const int*)  d_in[3];
    const float* norm  = (const float*)d_in[4];
    const float* emb_w = (const float*)d_in[5];
    const float* sm_w  = (const float*)d_in[6];
    const float* sm_b  = (const float*)d_in[7];
    const float* V1    = (const float*)d_in[8];
    const float* comp1 = (const float*)d_in[9];
    const float* b1    = (const float*)d_in[10];
    const float* V2    = (const float*)d_in[11];
    const float* comp2 = (const float*)d_in[12];
    const float* b2    = (const float*)d_in[13];
    float* out = (float*)d_out;

    // Workspace carve-up (floats / ints), ~58 MB total.
    float* Wt1 = (float*)d_ws;                             // 8*64*64 (transposed)
    float* Wt2 = Wt1 + NUM_RELS * H_DIM * H_DIM;           // 8*64*64 (transposed)
    float* x   = Wt2 + NUM_RELS * H_DIM * H_DIM;           // N*64
    float* h   = x + (long)N_NODES * H_DIM;                // N*64
    int* perm   = (int*)(h + (long)N_NODES * H_DIM);       // PERM_CAP
    int* counts = perm + PERM_CAP;                         // 8
    int* cursor = counts + NUM_RELS;                       // 8

    // Basis-combined relation weights (stored transposed for b64 B-fetch).
    compute_w_kernel<<<NUM_RELS, 256, 0, stream>>>(V1, comp1, Wt1);
    compute_w_kernel<<<NUM_RELS, 256, 0, stream>>>(V2, comp2, Wt2);

    // Embedding gather + size-matcher GEMM (WMMA).
    size_matcher_kernel<<<N_NODES / 16, 128, 0, stream>>>(feat, emb_w, sm_w, sm_b, x);

    // Bucket edges by relation with 16-padding.
    init_sort_kernel<<<(PERM_CAP + 255) / 256, 256, 0, stream>>>(perm, counts, PERM_CAP);
    hist_kernel<<<N_EDGES / 256, 256, 0, stream>>>(etype, counts);
    prefix_kernel<<<1, 1, 0, stream>>>(counts, cursor);
    scatter_kernel<<<N_EDGES / 256, 256, 0, stream>>>(etype, cursor, perm);

    // Layer 1: h = segsum((x[src] @ W1[r]) * norm) + b1   (ReLU deferred).
    init_bias_kernel<<<(N_NODES * H_DIM) / 256, 256, 0, stream>>>(h, b1);
    rgcn_layer_kernel<false><<<(NTILES + 3) / 4, 128, 0, stream>>>(
        x, Wt1, perm, src, dst, etype, norm, h, NTILES);

    // Layer 2: out = segsum((relu(h)[src] @ W2[r]) * norm) + b2.
    init_bias_kernel<<<(N_NODES * H_DIM) / 256, 256, 0, stream>>>(out, b2);
    rgcn_layer_kernel<true><<<(NTILES + 3) / 4, 128, 0, stream>>>(
        h, Wt2, perm, src, dst, etype, norm, out, NTILES);
}